// my_loss_18047452578173
// MI455X (gfx1250) — compile-verified
//
#include <hip/hip_runtime.h>

typedef __attribute__((ext_vector_type(2))) float v2f;
typedef __attribute__((ext_vector_type(8))) float v8f;

#define BATCH   16
#define SEQ     512
#define VOCAB   32000
#define NPOS    (BATCH * SEQ)   // 8192 gathered positions
#define NTHR    1024            // one workgroup = 32 wave32 waves

__global__ __launch_bounds__(NTHR)
void nll_masked_mean_kernel(const float* __restrict__ probs,
                            const int*   __restrict__ target,
                            float*       __restrict__ result)
{
    __shared__ float s_sum[NTHR];
    __shared__ float s_cnt[NTHR];
    __shared__ float s_fin[4];

    const int tid = threadIdx.x;

    // ---- Phase 1: gather + per-thread partial (8 independent loads in flight) ----
    float sum = 0.0f;
    float cnt = 0.0f;
    int   t[NPOS / NTHR];

    #pragma unroll
    for (int i = 0; i < NPOS / NTHR; ++i)
        t[i] = target[tid + i * NTHR];           // coalesced

    #pragma unroll
    for (int i = 0; i < NPOS / NTHR; ++i) {
        const int idx = tid + i * NTHR;
        if (t[i] != 0) {                          // MASK_IDX == 0
            float p = probs[(size_t)idx * VOCAB + (size_t)t[i]]; // scattered gather
            sum -= logf(p);
            cnt += 1.0f;
        }
    }

    s_sum[tid] = sum;
    s_cnt[tid] = cnt;
    __syncthreads();

    // ---- Phase 2: LDS tree reduction 1024 -> 64 ----
    #pragma unroll
    for (int stride = NTHR / 2; stride >= 64; stride >>= 1) {
        if (tid < stride) {
            s_sum[tid] += s_sum[tid + stride];
            s_cnt[tid] += s_cnt[tid + stride];
        }
        __syncthreads();
    }

    // ---- Phase 3: wave 0 folds the last 64 partials with V_WMMA_F32_16X16X4_F32 ----
    // A (16x4 f32) = the 64 partials (any bijection onto A works since B == all-ones):
    //   lane holds A-VGPR0 = s[lane], A-VGPR1 = s[lane+32].
    // With B = 1: D[m][n] = sum_k A[m][k]  (exact fp32 accumulation).
    // C/D layout: lane 0 VGPRs 0..7 = rows 0..7, lane 16 VGPRs 0..7 = rows 8..15.
    if (tid < 32) {                                // wave-uniform: EXEC all-1s for wave 0
        v2f a_s, a_c, ones;
        a_s.x = s_sum[tid];  a_s.y = s_sum[tid + 32];
        a_c.x = s_cnt[tid];  a_c.y = s_cnt[tid + 32];
        ones.x = 1.0f;       ones.y = 1.0f;
        v8f c0 = {};

        v8f d_s = __builtin_amdgcn_wmma_f32_16x16x4_f32(
            false, a_s, false, ones, (short)0, c0, false, false);
        v8f d_c = __builtin_amdgcn_wmma_f32_16x16x4_f32(
            false, a_c, false, ones, (short)0, c0, false, false);

        float ls = d_s[0] + d_s[1] + d_s[2] + d_s[3] +
                   d_s[4] + d_s[5] + d_s[6] + d_s[7];
        float lc = d_c[0] + d_c[1] + d_c[2] + d_c[3] +
                   d_c[4] + d_c[5] + d_c[6] + d_c[7];

        if (tid == 0)  { s_fin[0] = ls; s_fin[2] = lc; }   // rows 0..7
        if (tid == 16) { s_fin[1] = ls; s_fin[3] = lc; }   // rows 8..15
    }
    __syncthreads();

    if (tid == 0) {
        float total = s_fin[0] + s_fin[1];
        float count = s_fin[2] + s_fin[3];
        result[0] = total / count;
    }
}

extern "C" void kernel_launch(void* const* d_in, const int* in_sizes, int n_in,
                              void* d_out, int out_size, void* d_ws, size_t ws_size,
                              hipStream_t stream)
{
    (void)in_sizes; (void)n_in; (void)out_size; (void)d_ws; (void)ws_size;
    const float* probs  = (const float*)d_in[0];
    const int*   target = (const int*)d_in[1];
    float*       result = (float*)d_out;

    nll_masked_mean_kernel<<<dim3(1), dim3(NTHR), 0, stream>>>(probs, target, result);
}